// RPNLoss_51213190038050
// MI455X (gfx1250) — compile-verified
//
#include <hip/hip_runtime.h>
#include <hip/hip_bf16.h>

#define EPSF 1e-10f

typedef float v2f __attribute__((ext_vector_type(2)));
typedef float v8f __attribute__((ext_vector_type(8)));

// Wave32 cross-lane sum of two accumulators using one V_WMMA_F32_16X16X4_F32.
// A-VGPR0 = a0 (K=0 lanes0-15, K=2 lanes16-31), A-VGPR1 = a1 (K=1/K=3).
// B selects: B[k][n] = (k even ? n<8 : n>=8)  ->  b0 = ((lane&15)<8), b1 = 1-b0.
// After summing the 8 D VGPRs and adding the cross-half (xor 16) partner:
//   lanes with (lane&15)<8  hold sum_{l=0..31} a0(l)
//   lanes with (lane&15)>=8 hold sum_{l=0..31} a1(l)
__device__ __forceinline__ float wave_sum_pair_wmma(float a0, float a1, float b0, float b1) {
#if __has_builtin(__builtin_amdgcn_wmma_f32_16x16x4_f32)
    v2f A; A[0] = a0; A[1] = a1;
    v2f B; B[0] = b0; B[1] = b1;
    v8f C = {};
    C = __builtin_amdgcn_wmma_f32_16x16x4_f32(false, A, false, B, (short)0, C, false, false);
    float r = ((C[0] + C[1]) + (C[2] + C[3])) + ((C[4] + C[5]) + (C[6] + C[7]));
    r += __shfl_xor(r, 16, 32);
    return r;
#else
    // Fallback: plain shuffle reduction; replicate the same lane->value mapping.
    float s0 = a0, s1 = a1;
    for (int off = 16; off; off >>= 1) {
        s0 += __shfl_xor(s0, off, 32);
        s1 += __shfl_xor(s1, off, 32);
    }
    int lane = __lane_id();
    return ((lane & 15) < 8) ? s0 : s1;
#endif
}

__global__ void rpn_ws_init(float* ws_f, int* ws_i) {
    int t = threadIdx.x;
    if (t < 6) ws_f[t] = 0.0f;
    if (t == 6) ws_i[0] = -1;
}

__global__ __launch_bounds__(256) void rpn_reduce(
    const float* __restrict__ bbox_pred,   // [N,4]
    const float* __restrict__ class_logits,// [N,2]
    const float* __restrict__ sigma_pred,  // [N,4]
    const float* __restrict__ gt_bbox,     // [N,4]
    const int*   __restrict__ gt_label,    // [N]
    const int*   __restrict__ valid_idx,   // [V]
    float* ws_f, int* ws_i, int V)
{
    // acc: 0=class CE sum, 1=bbox sum (pos), 2=sigma sum (pos), 3=neg sum, 4=n_pos, 5=n_neg
    float acc0 = 0.f, acc1 = 0.f, acc2 = 0.f, acc3 = 0.f, acc4 = 0.f, acc5 = 0.f;
    int lastp = -1;

    const int stride = gridDim.x * blockDim.x;
    for (int g = blockIdx.x * blockDim.x + threadIdx.x; g < V; g += stride) {
        const size_t i = (size_t)valid_idx[g];
        const int lbl  = gt_label[i];

        // classification: -log_softmax(logits)[lbl], stable 2-class logsumexp
        const float2 lg = *(const float2*)(class_logits + 2u * i);
        const float m   = fmaxf(lg.x, lg.y);
        const float lse = m + __logf(__expf(lg.x - m) + __expf(lg.y - m));
        acc0 += lse - (lbl ? lg.y : lg.x);

        const float4 p = *(const float4*)(bbox_pred + 4u * i);
        const float4 t = *(const float4*)(gt_bbox   + 4u * i);
        const float4 s = *(const float4*)(sigma_pred + 4u * i);

        const float d0 = p.x - t.x, d1 = p.y - t.y, d2 = p.z - t.z, d3 = p.w - t.w;
        const float q0 = 0.5f * d0 * d0, q1 = 0.5f * d1 * d1;
        const float q2 = 0.5f * d2 * d2, q3 = 0.5f * d3 * d3;

        const float e0 = EPSF + s.x, e1 = EPSF + s.y, e2 = EPSF + s.z, e3 = EPSF + s.w;
        const float r0 = 1.0f / e0, r1 = 1.0f / e1, r2 = 1.0f / e2, r3 = 1.0f / e3;

        const float per_bbox  = (q0 * r0 + q1 * r1) + (q2 * r2 + q3 * r3);
        const float per_sigma = 0.5f * ((__logf(e0) + __logf(e1)) + (__logf(e2) + __logf(e3)));
        const float per_neg   = (r0 + r1) + (r2 + r3);

        if (lbl == 1) {
            acc1 += per_bbox;
            acc2 += per_sigma;
            acc4 += 1.0f;
            lastp = (g > lastp) ? g : lastp;
        } else {
            acc3 += per_neg;
            acc5 += 1.0f;
        }
    }

    // ---- in-wave reduction (WMMA), then cross-wave via LDS ----
    const int lane = threadIdx.x & 31;
    const int wave = threadIdx.x >> 5;
    const float b0 = ((lane & 15) < 8) ? 1.0f : 0.0f;
    const float b1 = 1.0f - b0;

    __shared__ float smem[8][6];
    __shared__ int   smax[8];

    float r01 = wave_sum_pair_wmma(acc0, acc1, b0, b1);
    float r23 = wave_sum_pair_wmma(acc2, acc3, b0, b1);
    float r45 = wave_sum_pair_wmma(acc4, acc5, b0, b1);
    if (lane == 0) { smem[wave][0] = r01; smem[wave][2] = r23; smem[wave][4] = r45; }
    if (lane == 8) { smem[wave][1] = r01; smem[wave][3] = r23; smem[wave][5] = r45; }

    for (int off = 16; off; off >>= 1) {
        int o = __shfl_xor(lastp, off, 32);
        lastp = (o > lastp) ? o : lastp;
    }
    if (lane == 0) smax[wave] = lastp;

    __syncthreads();

    if (threadIdx.x < 6) {
        float sum = 0.f;
        #pragma unroll
        for (int w = 0; w < 8; ++w) sum += smem[w][threadIdx.x];
        atomicAdd(&ws_f[threadIdx.x], sum);
    }
    if (threadIdx.x == 6) {
        int mx = -1;
        #pragma unroll
        for (int w = 0; w < 8; ++w) mx = (smax[w] > mx) ? smax[w] : mx;
        atomicMax(&ws_i[0], mx);
    }
}

__global__ void rpn_finalize(
    const float* __restrict__ bbox_pred,
    const float* __restrict__ gt_bbox,
    const int*   __restrict__ valid_idx,
    const float* ws_f, const int* ws_i,
    float* out, int V)
{
    if (threadIdx.x != 0 || blockIdx.x != 0) return;
    const float n_pos = ws_f[4];
    const float n_neg = ws_f[5];
    out[0] = ws_f[0] / (float)V;   // class_loss (mean CE)
    out[1] = ws_f[1] / n_pos;      // reg_loss_bbox
    out[2] = ws_f[2] / n_pos;      // reg_loss_sigma
    out[3] = ws_f[3] / n_neg;      // reg_loss_neg

    int lp = ws_i[0];
    if (lp < 0) lp = V - 1;        // mimic jnp index -1 wrap when no positives
    const size_t i = (size_t)valid_idx[lp];
    const float4 p = *(const float4*)(bbox_pred + 4u * i);
    const float4 t = *(const float4*)(gt_bbox   + 4u * i);
    const float d0 = p.x - t.x, d1 = p.y - t.y, d2 = p.z - t.z, d3 = p.w - t.w;
    const float sq = 0.5f * ((d0 * d0 + d1 * d1) + (d2 * d2 + d3 * d3));
    out[4] = sq / n_pos;           // reg_loss_bbox_only (last positive anchor)
}

extern "C" void kernel_launch(void* const* d_in, const int* in_sizes, int n_in,
                              void* d_out, int out_size, void* d_ws, size_t ws_size,
                              hipStream_t stream) {
    const float* bbox_pred    = (const float*)d_in[0];
    const float* class_logits = (const float*)d_in[1];
    const float* sigma_pred   = (const float*)d_in[2];
    const float* gt_bbox      = (const float*)d_in[3];
    const int*   gt_label     = (const int*)d_in[4];
    const int*   valid_idx    = (const int*)d_in[5];
    const int V = in_sizes[5];

    float* ws_f = (float*)d_ws;
    int*   ws_i = (int*)d_ws + 6;
    float* out  = (float*)d_out;

    rpn_ws_init<<<1, 32, 0, stream>>>(ws_f, ws_i);

    int blocks = (V + 255) / 256;
    if (blocks > 2048) blocks = 2048;
    if (blocks < 1) blocks = 1;
    rpn_reduce<<<blocks, 256, 0, stream>>>(bbox_pred, class_logits, sigma_pred, gt_bbox,
                                           gt_label, valid_idx, ws_f, ws_i, V);

    rpn_finalize<<<1, 32, 0, stream>>>(bbox_pred, gt_bbox, valid_idx, ws_f, ws_i, out, V);
}